// model_nn_37194416783788
// MI455X (gfx1250) — compile-verified
//
#include <hip/hip_runtime.h>

typedef __attribute__((ext_vector_type(16))) __bf16  v16bf;
typedef __attribute__((ext_vector_type(8)))  __bf16  v8bf;
typedef __attribute__((ext_vector_type(16))) float   v16f;
typedef __attribute__((ext_vector_type(8)))  unsigned v8u;
typedef __attribute__((ext_vector_type(4)))  unsigned v4u;

#define N_LAYERS 9

// f32 -> bf16 bits, round-to-nearest-even (host-side style helper, used only in
// the one-time LDS staging loops).
__device__ __forceinline__ unsigned short bf_bits(float f) {
    unsigned u = __builtin_bit_cast(unsigned, f);
    return (unsigned short)((u + 0x7FFFu + ((u >> 16) & 1u)) >> 16);
}
__device__ __forceinline__ __bf16 to_bf16(float f) {
    unsigned short h = bf_bits(f);
    __bf16 out;
    __builtin_memcpy(&out, &h, 2);
    return out;
}
__device__ __forceinline__ unsigned pack2_bf16(float lo, float hi) {
    return (unsigned)bf_bits(lo) | ((unsigned)bf_bits(hi) << 16);
}

// Packed bf16x2 leaky-ReLU: max(x, 0.01*x) on both halves.
// 0x3C24 = bf16(0.01) (RNE).
__device__ __forceinline__ unsigned lrelu_pk(unsigned u) {
    unsigned m, r;
    const unsigned c = 0x3C243C24u;
    asm("v_pk_mul_bf16 %0, %1, %2" : "=v"(m) : "v"(u), "v"(c));
    asm("v_pk_max_num_bf16 %0, %1, %2" : "=v"(r) : "v"(u), "v"(m));
    return r;
}

// K-map for the 16-bit A/B fragment layout (wave32):
//   element e (0..15), laneHi = lane>>4 :  K = ((e&8)<<1) + 8*laneHi + (e&7)
__device__ __forceinline__ int kmap(int lane, int e) {
    return ((e & 8) << 1) + ((lane >> 4) << 3) + (e & 7);
}

__global__ __launch_bounds__(256) void mlp_wmma_kernel(
    const float* __restrict__ x,
    const float* __restrict__ W_in,  const float* __restrict__ b_in,
    const float* __restrict__ W_hid, const float* __restrict__ b_hid,
    const float* __restrict__ W_out, const float* __restrict__ b_out,
    float* __restrict__ out, int n)
{
    // Pre-swizzled weight fragments, staged once per block (~31 KB LDS).
    __shared__ __align__(16) __bf16   s_afrag[N_LAYERS][2][32][16]; // A = rows of W^T (bf16 frag order)
    __shared__ __align__(16) __bf16   s_cfrag[N_LAYERS][2][32][8];  // bias in bf16 C/D layout
    __shared__ __align__(16) unsigned s_winpk[32][8];               // W_in  packed bf16x2, frag K order
    __shared__ __align__(16) unsigned s_binpk[32][8];               // b_in  packed bf16x2, frag K order
    __shared__ __align__(16) float    s_wout [32][16];              // W_out f32, frag K order

    const int tid = threadIdx.x;

    // ---- one-time fragment staging ----
    for (int idx = tid; idx < N_LAYERS * 2 * 32 * 16; idx += blockDim.x) {
        int e = idx & 15, lane = (idx >> 4) & 31, a = (idx >> 9) & 1, l = idx >> 10;
        int K = kmap(lane, e);
        int m = a * 16 + (lane & 15);
        s_afrag[l][a][lane][e] = to_bf16(W_hid[l * 1024 + K * 32 + m]);   // W^T[a*16+m][K]
    }
    for (int idx = tid; idx < N_LAYERS * 2 * 32 * 8; idx += blockDim.x) {
        int e = idx & 7, lane = (idx >> 3) & 31, a = (idx >> 8) & 1, l = idx >> 9;
        int m = a * 16 + ((lane >> 4) << 3) + e;   // 16-bit C/D: M = e + 8*laneHi
        s_cfrag[l][a][lane][e] = to_bf16(b_hid[l * 32 + m]);
    }
    for (int idx = tid; idx < 32 * 8; idx += blockDim.x) {
        int p = idx & 7, lane = idx >> 3;
        int K0 = kmap(lane, 2 * p), K1 = kmap(lane, 2 * p + 1);
        s_winpk[lane][p] = pack2_bf16(W_in[K0], W_in[K1]);
        s_binpk[lane][p] = pack2_bf16(b_in[K0], b_in[K1]);
    }
    for (int idx = tid; idx < 32 * 16; idx += blockDim.x) {
        int e = idx & 15, lane = idx >> 4;
        s_wout[lane][e] = W_out[kmap(lane, e)];
    }
    __syncthreads();

    const int lane = tid & 31;
    const int col  = lane & 15;
    const int hi   = lane >> 4;
    const int wave = (blockIdx.x * blockDim.x + tid) >> 5;
    const int totalWaves = (gridDim.x * blockDim.x) >> 5;
    const int nChunks = (n + 63) >> 6;          // 64 samples (4 tiles of 16) per wave-chunk
    const float bo = b_out[0];

    // Loop-invariant operands held in registers.
    const v8u  winr  = *(const v8u*) &s_winpk[lane][0];
    const v8u  binr  = *(const v8u*) &s_binpk[lane][0];
    const v16f woutr = *(const v16f*)&s_wout [lane][0];

    for (int chunk = wave; chunk < nChunks; chunk += totalWaves) {
        const int base = chunk << 6;
        v16bf B[4];   // activations^T as bf16 B fragments (32 features x 16 samples)

        // ---- input layer: h0 = lrelu(x*W_in + b_in), all in packed bf16 ----
        #pragma unroll
        for (int t = 0; t < 4; ++t) {
            int s = base + t * 16 + col;
            if (s >= n) s = n - 1;               // clamp keeps EXEC all-ones for WMMA
            float xv = x[s];
            unsigned xb  = __builtin_bit_cast(unsigned, xv);
            unsigned xpk = __builtin_amdgcn_perm(xb, xb, 0x07060302);  // {bf16(x), bf16(x)}
            v8u bt;
            #pragma unroll
            for (int p = 0; p < 8; ++p) {
                unsigned h;
                asm("v_pk_fma_bf16 %0, %1, %2, %3"
                    : "=v"(h) : "v"(xpk), "v"(winr[p]), "v"(binr[p]));
                bt[p] = lrelu_pk(h);
            }
            B[t] = __builtin_bit_cast(v16bf, bt);
        }

        // ---- 9 hidden layers: bf16-out WMMA, D concat feeds next B directly ----
        #pragma unroll 1
        for (int l = 0; l < N_LAYERS; ++l) {
            v16bf A1 = *(const v16bf*)&s_afrag[l][0][lane][0];
            v16bf A2 = *(const v16bf*)&s_afrag[l][1][lane][0];
            v8bf  C1 = *(const v8bf*) &s_cfrag[l][0][lane][0];
            v8bf  C2 = *(const v8bf*) &s_cfrag[l][1][lane][0];
            #pragma unroll
            for (int t = 0; t < 4; ++t) {
                v8bf D1 = __builtin_amdgcn_wmma_bf16_16x16x32_bf16(
                              false, A1, false, B[t], (short)0, C1, false, false);
                v8bf D2 = __builtin_amdgcn_wmma_bf16_16x16x32_bf16(
                              false, A2, false, B[t], (short)0, C2, false, false);
                v4u u1 = __builtin_bit_cast(v4u, D1);
                v4u u2 = __builtin_bit_cast(v4u, D2);
                v8u nb;
                #pragma unroll
                for (int p = 0; p < 4; ++p) {
                    nb[p]     = lrelu_pk(u1[p]);
                    nb[p + 4] = lrelu_pk(u2[p]);
                }
                B[t] = __builtin_bit_cast(v16bf, nb);   // pure register concat
            }
        }

        // ---- output layer: f32 dot(h, W_out) per sample across lane pair ----
        #pragma unroll
        for (int t = 0; t < 4; ++t) {
            v8u bu = __builtin_bit_cast(v8u, B[t]);
            float p = 0.f;
            #pragma unroll
            for (int q = 0; q < 8; ++q) {
                float lo = __builtin_bit_cast(float, bu[q] << 16);
                float hh = __builtin_bit_cast(float, bu[q] & 0xFFFF0000u);
                p = fmaf(lo, woutr[2 * q],     p);
                p = fmaf(hh, woutr[2 * q + 1], p);
            }
            p += __shfl_xor(p, 16, 32);          // combine K halves across lane pair
            int s = base + t * 16 + col;
            if (hi == 0 && s < n)
                out[s] = p + bo;
        }
    }
}

extern "C" void kernel_launch(void* const* d_in, const int* in_sizes, int n_in,
                              void* d_out, int out_size, void* d_ws, size_t ws_size,
                              hipStream_t stream) {
    const float* x     = (const float*)d_in[0];
    const float* W_in  = (const float*)d_in[1];
    const float* b_in  = (const float*)d_in[2];
    const float* W_hid = (const float*)d_in[3];
    const float* b_hid = (const float*)d_in[4];
    const float* W_out = (const float*)d_in[5];
    const float* b_out = (const float*)d_in[6];
    float* out = (float*)d_out;

    int n = in_sizes[0];
    int chunks = (n + 63) / 64;
    int blocks = (chunks + 7) / 8;   // 8 waves per 256-thread block
    if (blocks > 1024) blocks = 1024;
    if (blocks < 1) blocks = 1;

    mlp_wmma_kernel<<<blocks, 256, 0, stream>>>(x, W_in, b_in, W_hid, b_hid,
                                                W_out, b_out, out, n);
}